// CRFs_89988154786459
// MI455X (gfx1250) — compile-verified
//
#include <hip/hip_runtime.h>
#include <hip/hip_bf16.h>
#include <math.h>

typedef __attribute__((ext_vector_type(2))) float v2f;
typedef __attribute__((ext_vector_type(4))) float v4f;
typedef __attribute__((ext_vector_type(8))) float v8f;

#define BB 64
#define TT 512
#define DD 768
#define MROWS (BB * TT)          // 32768
#define NTILES (MROWS / 16)      // 2048

// ---------------------------------------------------------------------------
// Pack W1 (768x2) / W2 (768x9) into a b128-per-lane WMMA-B layout.
// Wq is indexed [g][lane][e] (g = K/8 group, 96 groups):
//   col = lane&15, half = lane>>4, k = g*8 + half*4 + e
//   Wq[(g*32+lane)*4+e] = Wp[k][col],  Wp[k][c] = c<2 ? W1[k][c]
//                                              : c<11 ? W2[k][c-2] : 0
// With the A fragment built from a float4 at x[row, g*8 + 4*half], the K-slot
// permutation of A and B agree, so each WMMA pair needs 1 b128 load per side.
// ---------------------------------------------------------------------------
__global__ void build_wq_kernel(const float* __restrict__ W1,
                                const float* __restrict__ W2,
                                float* __restrict__ Wq) {
    int i = blockIdx.x * blockDim.x + threadIdx.x;   // 0 .. 96*32*4-1
    if (i >= 96 * 32 * 4) return;
    int e    = i & 3;
    int lane = (i >> 2) & 31;
    int g    = i >> 7;
    int col  = lane & 15;
    int half = lane >> 4;
    int k    = g * 8 + half * 4 + e;
    float v  = (col < 2) ? W1[k * 2 + col]
             : (col < 11) ? W2[k * 9 + (col - 2)] : 0.0f;
    Wq[i] = v;
}

// ---------------------------------------------------------------------------
// FP32 WMMA GEMM: (32768 x 768) x (768 x 16pad) -> out1 (.,2), out2 (.,9)
// One wave per 16-row tile; exactly 2048 waves, EXEC all-ones everywhere.
// Per 8-K step: one b128 A-load + one b128 B-load + two v_wmma_f32_16x16x4_f32.
// ---------------------------------------------------------------------------
__global__ __launch_bounds__(256) void gemm_wmma_kernel(
    const float* __restrict__ x, const float* __restrict__ Wq,
    const float* __restrict__ b1, const float* __restrict__ b2,
    float* __restrict__ out1, float* __restrict__ out2) {
    const int lane = threadIdx.x & 31;
    const int wave = blockIdx.x * (blockDim.x >> 5) + (threadIdx.x >> 5);
    const int row0 = wave * 16;
    const int half = lane >> 4;         // 0 or 1
    const int l15  = lane & 15;
    const long arow = (long)(row0 + l15) * DD;
    const v4f* __restrict__ wq4 = (const v4f*)Wq;

    v8f c = {};
    for (int k0 = 0; k0 < DD; k0 += 32) {
        __builtin_prefetch(x + arow + k0 + 256, 0, 1);   // stream ~1KB ahead
#pragma unroll
        for (int k = k0; k < k0 + 32; k += 8) {
            v4f qa = *(const v4f*)(x + arow + k + 4 * half);
            v4f qb = wq4[(k >> 3) * 32 + lane];
            v2f a0 = {qa.x, qa.y}, a1 = {qa.z, qa.w};
            v2f bb0 = {qb.x, qb.y}, bb1 = {qb.z, qb.w};
            c = __builtin_amdgcn_wmma_f32_16x16x4_f32(
                    false, a0, false, bb0, (short)0, c, false, false);
            c = __builtin_amdgcn_wmma_f32_16x16x4_f32(
                    false, a1, false, bb1, (short)0, c, false, false);
        }
    }

    // C/D layout: VGPR v holds M = v + 8*half, N = l15
    const int col = l15;
    float bias = (col < 2) ? b1[col] : ((col < 11) ? b2[col - 2] : 0.0f);
#pragma unroll
    for (int v = 0; v < 8; ++v) {
        int row = row0 + v + half * 8;
        float val = c[v] + bias;
        if (col < 2)       out1[row * 2 + col]       = val;
        else if (col < 11) out2[row * 9 + (col - 2)] = val;
    }
}

// ---------------------------------------------------------------------------
// CRF: one workgroup per batch element, two independent waves.
//   wave 0: logsumexp forward + gold-path unary/binary score  -> ll[b]
//   wave 1: Viterbi forward (backpointers in LDS) + backtrace -> vit[b][t]
// Lane j owns alpha[j] and trans[:,j] (registers); the reduction over previous
// labels broadcasts alpha via __shfl with compile-time lane index.
// ---------------------------------------------------------------------------
template <int L, int WHICH>
__global__ __launch_bounds__(64) void crf_kernel(
    const float* __restrict__ logits, const float* __restrict__ trans,
    const int* __restrict__ label, const int* __restrict__ seqlen,
    float* __restrict__ ll, int* __restrict__ vit) {
    __shared__ float s_trans[L * L];
    __shared__ unsigned char s_bp[TT * L];
    const int b    = blockIdx.x;
    const int tid  = threadIdx.x;
    const int wv   = tid >> 5;          // 0 = logsumexp, 1 = viterbi
    const int lane = tid & 31;
    for (int i = tid; i < L * L; i += 64) s_trans[i] = trans[i];
    __syncthreads();

    const int slen = seqlen[b];
    const float* lg = logits + (long)b * TT * L;
    const int jj = (lane < L) ? lane : (L - 1);   // clamp idle lanes

    float trc[L];                        // trans[:, j] column
#pragma unroll
    for (int i = 0; i < L; ++i) trc[i] = s_trans[i * L + jj];
    float a = lg[jj];                    // alpha[j] at t = 0
    if (lane >= L) a = -1e30f;

    if (wv == 0) {
        // ---------------- log-likelihood wave ----------------
        int tagp = label[b * TT];
        if (WHICH == 0) tagp = (tagp > 0) ? 1 : 0;
        float unary  = __shfl(a, tagp);  // logits[b,0,tag0]; mask[0] always 1
        float binary = 0.0f;
        for (int t = 1; t < TT; ++t) {
            const bool m = (t < slen);
            float lt = lg[t * L + jj];
            float vv[L];
#pragma unroll
            for (int i = 0; i < L; ++i) vv[i] = __shfl(a, i) + trc[i];
            float mx = vv[0];
#pragma unroll
            for (int i = 1; i < L; ++i) mx = fmaxf(mx, vv[i]);
            float s = 0.0f;
#pragma unroll
            for (int i = 0; i < L; ++i) s += __expf(vv[i] - mx);
            float nl = mx + __logf(s) + lt;
            if (m) a = nl;
            int tag = label[b * TT + t];
            if (WHICH == 0) tag = (tag > 0) ? 1 : 0;
            float ltt = __shfl(lt, tag);
            if (m) { unary += ltt; binary += s_trans[tagp * L + tag]; }
            tagp = tag;
        }
        float av[L];
#pragma unroll
        for (int i = 0; i < L; ++i) av[i] = __shfl(a, i);
        float mx = av[0];
#pragma unroll
        for (int i = 1; i < L; ++i) mx = fmaxf(mx, av[i]);
        float s = 0.0f;
#pragma unroll
        for (int i = 0; i < L; ++i) s += __expf(av[i] - mx);
        if (lane == 0) ll[b] = unary + binary - (mx + __logf(s));
    } else {
        // ---------------- viterbi wave ----------------
        for (int t = 1; t < TT; ++t) {
            const bool m = (t < slen);
            float lt = lg[t * L + jj];
            float best = -INFINITY; int bi = 0;
#pragma unroll
            for (int i = 0; i < L; ++i) {
                float v = __shfl(a, i) + trc[i];
                if (v > best) { best = v; bi = i; }   // first max wins
            }
            float nv = best + lt;
            int bp = m ? bi : jj;                     // identity when masked
            if (lane < L) s_bp[t * L + lane] = (unsigned char)bp;
            if (m) a = nv;
        }
        float bmax = -INFINITY; int last = 0;
#pragma unroll
        for (int i = 0; i < L; ++i) {
            float v = __shfl(a, i);
            if (v > bmax) { bmax = v; last = i; }
        }
        if (lane == 0) {
            int* vb = vit + b * TT;
            int tag = last;
            vb[TT - 1] = tag;
            for (int t = TT - 1; t >= 1; --t) {       // LDS pointer chase
                tag = s_bp[t * L + tag];
                vb[t - 1] = tag;
            }
        }
    }
}

// ---------------------------------------------------------------------------
// Final reduction: viterbi merge, accuracies, loss. Single 256-thread block.
// ---------------------------------------------------------------------------
__device__ float block_reduce_sum(float v, float* sh) {
    int tid = threadIdx.x;
    sh[tid] = v;
    __syncthreads();
    for (int s = 128; s > 0; s >>= 1) {
        if (tid < s) sh[tid] += sh[tid + s];
        __syncthreads();
    }
    float r = sh[0];
    __syncthreads();
    return r;
}

__global__ __launch_bounds__(256) void finalize_kernel(
    const int* __restrict__ label, const int* __restrict__ seqlen,
    const float* __restrict__ ll1, const float* __restrict__ ll2,
    const int* __restrict__ vit1, const int* __restrict__ vit2,
    float* __restrict__ d_out) {
    __shared__ float sh[256];
    const int tid = threadIdx.x;
    float s1 = 0.0f, s2 = 0.0f, sa = 0.0f;
    for (int idx = tid; idx < MROWS; idx += blockDim.x) {
        int b = idx >> 9, t = idx & 511;
        bool m = (t < seqlen[b]);
        int lab  = label[idx];
        int lab1 = (lab > 0) ? 1 : 0;
        int v1 = vit1[idx], v2 = vit2[idx];
        int v  = (v1 == 0) ? 0 : v2;
        d_out[idx] = (float)v;
        if (m) {
            s1 += (v1 == lab1) ? 1.0f : 0.0f;
            s2 += (v2 == lab)  ? 1.0f : 0.0f;
            sa += (v  == lab)  ? 1.0f : 0.0f;
        }
    }
    float l1 = 0.0f, l2 = 0.0f, tot = 0.0f;
    if (tid < BB) {
        l1  = -ll1[tid];
        l2  = -ll2[tid];
        tot = (float)seqlen[tid];
    }
    float S1  = block_reduce_sum(s1, sh);
    float S2  = block_reduce_sum(s2, sh);
    float SA  = block_reduce_sum(sa, sh);
    float L1s = block_reduce_sum(l1, sh);
    float L2s = block_reduce_sum(l2, sh);
    float TOT = block_reduce_sum(tot, sh);
    if (tid == 0) {
        float loss1 = L1s / (float)BB;
        float loss2 = L2s / (float)BB;
        d_out[MROWS + 0] = (loss1 + 8.0f * loss2) / 9.0f;
        d_out[MROWS + 1] = S1 / TOT;
        d_out[MROWS + 2] = S2 / TOT;
        d_out[MROWS + 3] = SA / TOT;
    }
}

// ---------------------------------------------------------------------------
extern "C" void kernel_launch(void* const* d_in, const int* in_sizes, int n_in,
                              void* d_out, int out_size, void* d_ws, size_t ws_size,
                              hipStream_t stream) {
    const float* x      = (const float*)d_in[0];
    const int*   label  = (const int*)d_in[1];
    const int*   seqlen = (const int*)d_in[2];
    const float* W1     = (const float*)d_in[3];
    const float* b1     = (const float*)d_in[4];
    const float* W2     = (const float*)d_in[5];
    const float* b2     = (const float*)d_in[6];
    const float* trans1 = (const float*)d_in[7];
    const float* trans2 = (const float*)d_in[8];

    // workspace bump allocator (16B aligned sections), ~1.8 MB total
    char* w = (char*)d_ws;
    auto alloc = [&](size_t bytes) -> char* {
        char* p = w;
        w += (bytes + 15) & ~(size_t)15;
        return p;
    };
    float* out1 = (float*)alloc((size_t)MROWS * 2 * 4);
    float* out2 = (float*)alloc((size_t)MROWS * 9 * 4);
    float* Wq   = (float*)alloc((size_t)96 * 32 * 4 * 4);
    float* ll1  = (float*)alloc((size_t)BB * 4);
    float* ll2  = (float*)alloc((size_t)BB * 4);
    int*   vit1 = (int*)alloc((size_t)MROWS * 4);
    int*   vit2 = (int*)alloc((size_t)MROWS * 4);

    build_wq_kernel<<<48, 256, 0, stream>>>(W1, W2, Wq);
    gemm_wmma_kernel<<<NTILES / 8, 256, 0, stream>>>(x, Wq, b1, b2, out1, out2);
    crf_kernel<2, 0><<<BB, 64, 0, stream>>>(out1, trans1, label, seqlen, ll1, vit1);
    crf_kernel<9, 1><<<BB, 64, 0, stream>>>(out2, trans2, label, seqlen, ll2, vit2);
    finalize_kernel<<<1, 256, 0, stream>>>(label, seqlen, ll1, ll2, vit1, vit2,
                                           (float*)d_out);
}